// Pytorch_filter_smoother_Obj_1400159338697
// MI455X (gfx1250) — compile-verified
//
#include <hip/hip_runtime.h>

typedef __attribute__((ext_vector_type(2))) float v2f;
typedef __attribute__((ext_vector_type(4))) float v4f;
typedef __attribute__((ext_vector_type(8))) float v8f;

#define WMMA4(A, B, C) \
  __builtin_amdgcn_wmma_f32_16x16x4_f32(false, (A), false, (B), (short)0, (C), false, false)

constexpr int kN = 2048;
constexpr int kB = 256;
constexpr int kD = 32;
constexpr size_t kStep = (size_t)kB * kD;  // floats per time index

// ---------------------------------------------------------------------------
// State-row permutation sigma: B-operand chunk c, lane-half h, vgpr v carries
// state row  colbase(c,h)+v.  Chosen so the B operand for the next step is a
// pure register re-pairing of the current step's C-layout WMMA result
// (no cross-lane / DS traffic on the serial critical path).  The permutation
// is compensated by permuting the A-matrix *columns* at (one-time) load.
// ---------------------------------------------------------------------------
__device__ __forceinline__ int colbase(int c, int half) {
  return 2 * c + 8 * half + (c >= 4 ? 8 : 0);
}

struct ATiles { v2f a[2][8]; };

// Load a 32x32 row-major matrix into A-operand tiles with sigma-permuted
// columns.  a[rt][c]: output rows 16rt..16rt+15, local K 0..3 mapping to
// global columns colbase(c,0)+{0,1}, colbase(c,1)+{0,1}.
// A 16x4 f32 layout: vgpr v: lanes 0-15 = (M=lane, K=v), lanes 16-31 = (M, K=v+2).
__device__ __forceinline__ void load_a_tiles(const float* __restrict__ M, ATiles& t, int lane) {
  const int r0 = lane & 15;
  const int half = lane >> 4;
#pragma unroll
  for (int rt = 0; rt < 2; ++rt) {
#pragma unroll
    for (int c = 0; c < 8; ++c) {
      const float* p = M + (16 * rt + r0) * kD + colbase(c, half);
      t.a[rt][c] = *(const v2f*)p;
    }
  }
}

// Load 16 contiguous 32-float column vectors into sigma-permuted B tiles.
__device__ __forceinline__ void load_b_cols(const float* __restrict__ src, v2f xb[8], int lane) {
  const int col = lane & 15;
  const int half = lane >> 4;
  const float* p = src + col * kD;
#pragma unroll
  for (int c = 0; c < 8; ++c) xb[c] = *(const v2f*)(p + colbase(c, half));
}

__device__ __forceinline__ void store_b_cols(float* __restrict__ dst, const v2f xb[8], int lane) {
  const int col = lane & 15;
  const int half = lane >> 4;
  float* p = dst + col * kD;
#pragma unroll
  for (int c = 0; c < 8; ++c) *(v2f*)(p + colbase(c, half)) = xb[c];
}

// C/D 16x16 f32 layout: vgpr v: lanes 0-15 = (M=v, N=lane), lanes 16-31 = (M=v+8, N).
// c0 = rows 0..15 tile, c1 = rows 16..31 tile, natural (unpermuted) rows.
__device__ __forceinline__ void load_c_cols(const float* __restrict__ src, v8f& c0, v8f& c1, int lane) {
  const int col = lane & 15;
  const int half = lane >> 4;
  const float* p = src + col * kD + 8 * half;
  v4f a = *(const v4f*)(p + 0);
  v4f b = *(const v4f*)(p + 4);
  v4f c = *(const v4f*)(p + 16);
  v4f d = *(const v4f*)(p + 20);
  c0 = (v8f){a[0], a[1], a[2], a[3], b[0], b[1], b[2], b[3]};
  c1 = (v8f){c[0], c[1], c[2], c[3], d[0], d[1], d[2], d[3]};
}

__device__ __forceinline__ void store_c_cols(float* __restrict__ dst, v8f c0, v8f c1, int lane) {
  const int col = lane & 15;
  const int half = lane >> 4;
  float* p = dst + col * kD + 8 * half;
  *(v4f*)(p + 0)  = (v4f){c0[0], c0[1], c0[2], c0[3]};
  *(v4f*)(p + 4)  = (v4f){c0[4], c0[5], c0[6], c0[7]};
  *(v4f*)(p + 16) = (v4f){c1[0], c1[1], c1[2], c1[3]};
  *(v4f*)(p + 20) = (v4f){c1[4], c1[5], c1[6], c1[7]};
}

// C-layout result -> sigma-permuted B operand: pure per-lane register re-pairing.
// Chunk c<4 needs rows 2c+8h+{0,1} = acc0[2c],acc0[2c+1]; c>=4 -> acc1[...].
__device__ __forceinline__ void c_to_b_regs(const v8f& c0, const v8f& c1, v2f xb[8]) {
#pragma unroll
  for (int c = 0; c < 4; ++c) {
    xb[c]     = (v2f){c0[2 * c], c0[2 * c + 1]};
    xb[c + 4] = (v2f){c1[2 * c], c1[2 * c + 1]};
  }
}

// One recurrence step:  D = Ain*inb + Axx*xb, 4 independent WMMA chains.
// The input-matmul (recurrence-independent) WMMAs seed each chain so the
// x-dependent critical path is only 4 WMMAs deep per chain.
__device__ __forceinline__ void step_mm(const ATiles& Axx, const v2f xb[8],
                                        const ATiles& Ain, const v2f inb[8],
                                        v8f& d0, v8f& d1) {
  v8f a0e = {0.f, 0.f, 0.f, 0.f, 0.f, 0.f, 0.f, 0.f};
  v8f a0o = a0e, a1e = a0e, a1o = a0e;
#pragma unroll
  for (int c = 0; c < 8; c += 2) {
    a0e = WMMA4(Ain.a[0][c], inb[c], a0e);
    a0o = WMMA4(Ain.a[0][c + 1], inb[c + 1], a0o);
    a1e = WMMA4(Ain.a[1][c], inb[c], a1e);
    a1o = WMMA4(Ain.a[1][c + 1], inb[c + 1], a1o);
  }
#pragma unroll
  for (int c = 0; c < 8; c += 2) {
    a0e = WMMA4(Axx.a[0][c], xb[c], a0e);
    a0o = WMMA4(Axx.a[0][c + 1], xb[c + 1], a0o);
    a1e = WMMA4(Axx.a[1][c], xb[c], a1e);
    a1o = WMMA4(Axx.a[1][c + 1], xb[c + 1], a1o);
  }
  d0 = a0e + a0o;
  d1 = a1e + a1o;
}

// Forward filter: hat[0] = x0; hat[n+1] = F*hat[n] + K*z[n].
// One wave per 16-batch column slice; state lives in WMMA registers.
__global__ void __launch_bounds__(32)
kf_forward_kernel(const float* __restrict__ z, const float* __restrict__ x0,
                  const float* __restrict__ F, const float* __restrict__ K,
                  float* __restrict__ hat) {
  const int lane = threadIdx.x & 31;
  const int b0 = blockIdx.x * 16;

  ATiles Fa, Ka;
  load_a_tiles(F, Fa, lane);
  load_a_tiles(K, Ka, lane);

  v2f xb[8];
  load_b_cols(x0 + (size_t)b0 * kD, xb, lane);
  store_b_cols(hat + (size_t)b0 * kD, xb, lane);  // hat[0] = init

  for (int n = 0; n < kN - 1; ++n) {
    const float* zsrc = z + (size_t)n * kStep + (size_t)b0 * kD;
    if (n + 6 < kN - 1)
      __builtin_prefetch(z + (size_t)(n + 6) * kStep + (size_t)b0 * kD + lane * 16, 0, 0);
    v2f zb[8];
    load_b_cols(zsrc, zb, lane);

    v8f d0, d1;
    step_mm(Fa, xb, Ka, zb, d0, d1);

    store_c_cols(hat + (size_t)(n + 1) * kStep + (size_t)b0 * kD, d0, d1, lane);
    c_to_b_regs(d0, d1, xb);
  }
}

// Backward smoother:
//   sN = hat[N-1] + Ka0*(z[N-1] - Ht*hat[N-1])
//   s[k] = Gr*s[k+1] + Gs*hat[k+1],  k = N-2..0
__global__ void __launch_bounds__(32)
kf_backward_kernel(const float* __restrict__ z, const float* __restrict__ Ht,
                   const float* __restrict__ Ka0, const float* __restrict__ Gs,
                   const float* __restrict__ Gr, const float* __restrict__ hat,
                   float* __restrict__ out2) {
  const int lane = threadIdx.x & 31;
  const int b0 = blockIdx.x * 16;
  const size_t last = (size_t)(kN - 1) * kStep + (size_t)b0 * kD;

  ATiles A1, A2;
  v2f sb[8];

  {  // prologue: sN
    load_a_tiles(Ht, A1, lane);
    v2f hb[8];
    load_b_cols(hat + last, hb, lane);
    v8f t0 = {0.f, 0.f, 0.f, 0.f, 0.f, 0.f, 0.f, 0.f};
    v8f t1 = t0;
#pragma unroll
    for (int c = 0; c < 8; ++c) t0 = WMMA4(A1.a[0][c], hb[c], t0);  // Ht*hat
#pragma unroll
    for (int c = 0; c < 8; ++c) t1 = WMMA4(A1.a[1][c], hb[c], t1);

    v8f zc0, zc1;
    load_c_cols(z + last, zc0, zc1, lane);
    v8f bz0 = zc0 - t0;  // bar_z, C layout
    v8f bz1 = zc1 - t1;
    v2f bb[8];
    c_to_b_regs(bz0, bz1, bb);

    load_a_tiles(Ka0, A2, lane);
    v8f s0, s1;
    load_c_cols(hat + last, s0, s1, lane);  // C seed = hat[N-1]
#pragma unroll
    for (int c = 0; c < 8; ++c) s0 = WMMA4(A2.a[0][c], bb[c], s0);  // + Ka0*bar_z
#pragma unroll
    for (int c = 0; c < 8; ++c) s1 = WMMA4(A2.a[1][c], bb[c], s1);

    store_c_cols(out2 + last, s0, s1, lane);
    c_to_b_regs(s0, s1, sb);
  }

  load_a_tiles(Gr, A1, lane);
  load_a_tiles(Gs, A2, lane);

  for (int k = kN - 2; k >= 0; --k) {
    const float* hsrc = hat + (size_t)(k + 1) * kStep + (size_t)b0 * kD;
    if (k - 6 >= 0)
      __builtin_prefetch(hat + (size_t)(k - 5) * kStep + (size_t)b0 * kD + lane * 16, 0, 0);
    v2f hb[8];
    load_b_cols(hsrc, hb, lane);

    v8f d0, d1;
    step_mm(A1, sb, A2, hb, d0, d1);  // Gr*s[k+1] + Gs*hat[k+1]

    store_c_cols(out2 + (size_t)k * kStep + (size_t)b0 * kD, d0, d1, lane);
    c_to_b_regs(d0, d1, sb);
  }
}

extern "C" void kernel_launch(void* const* d_in, const int* in_sizes, int n_in,
                              void* d_out, int out_size, void* d_ws, size_t ws_size,
                              hipStream_t stream) {
  (void)in_sizes; (void)n_in; (void)d_ws; (void)ws_size; (void)out_size;
  const float* z   = (const float*)d_in[0];  // [N,B,32,1]
  const float* x0  = (const float*)d_in[1];  // [B,32,1]
  const float* F   = (const float*)d_in[2];  // tildeF 32x32
  const float* K   = (const float*)d_in[3];  // 32x32
  const float* Ht  = (const float*)d_in[4];  // H_transpose 32x32
  const float* Ka0 = (const float*)d_in[5];  // 32x32
  const float* Gs  = (const float*)d_in[6];  // smootherGain 32x32
  const float* Gr  = (const float*)d_in[7];  // smootherRecursiveGain 32x32

  float* hat  = (float*)d_out;               // hat_x       [N,B,32]
  float* out2 = hat + (size_t)kN * kB * kD;  // hat_x_given_N

  dim3 grid(kB / 16);
  dim3 block(32);
  kf_forward_kernel<<<grid, block, 0, stream>>>(z, x0, F, K, hat);
  kf_backward_kernel<<<grid, block, 0, stream>>>(z, Ht, Ka0, Gs, Gr, hat, out2);
}